// ParametricEqualizer_75866302316883
// MI455X (gfx1250) — compile-verified
//
#include <hip/hip_runtime.h>

#define NFILT   10
#define LSIG    65536
#define BLK     128            // block length N
#define NBLK    (LSIG/BLK)     // 512 blocks per channel
#define STATE   20             // cascade state dimension (2 per biquad)
#define CH_STRIDE 6144         // per-channel float stride in workspace
#define HPAD    16             // tail zero-padding for triangular edge
// workspace layout (floats, per channel)
#define OFF_H 0                              // h[BLK] impulse response
#define OFF_G (OFF_H + BLK)                  // G[STATE][BLK]
#define OFF_O (OFF_G + STATE*BLK)            // O[BLK][STATE]
#define OFF_T (OFF_O + BLK*STATE)            // T[STATE][STATE] = A^BLK

typedef float v2f __attribute__((ext_vector_type(2)));
typedef float v8f __attribute__((ext_vector_type(8)));

// ---------------------------------------------------------------------------
// Kernel 1: per-channel coefficients + block state-space tables via exact
// simulation of the reference DF2T cascade (impulse + 20 unit-state runs).
// ---------------------------------------------------------------------------
__global__ void eq_setup(const float* __restrict__ w0_in,
                         const float* __restrict__ qinv_in,
                         const float* __restrict__ lg_in,
                         float* __restrict__ ws) {
  const int ch = blockIdx.x;
  const int lane = threadIdx.x;
  __shared__ float cf[5][NFILT];

  if (lane < NFILT) {
    float W = w0_in[ch * NFILT + lane];
    float Q = qinv_in[ch * NFILT + lane];
    float Gg = lg_in[ch * NFILT + lane];
    float w  = 3.14159265358979323846f / (1.0f + expf(-W));   // pi*sigmoid
    float qi = expf(Q);
    float A  = expf(Gg);
    float cw = cosf(w);
    float al = 0.5f * sinf(w) * qi;
    float B0, B1, B2, A0, A1, A2;
    if (lane == 0) {                 // low shelf
      float Ap1 = A + 1.f, Am1 = A - 1.f, tsa = 2.f * sqrtf(A) * al;
      B0 = A * (Ap1 - Am1 * cw + tsa);
      B1 = 2.f * A * (Am1 - Ap1 * cw);
      B2 = A * (Ap1 - Am1 * cw - tsa);
      A0 = Ap1 + Am1 * cw + tsa;
      A1 = -2.f * (Am1 + Ap1 * cw);
      A2 = Ap1 + Am1 * cw - tsa;
    } else if (lane == NFILT - 1) {  // high shelf
      float Ap1 = A + 1.f, Am1 = A - 1.f, tsa = 2.f * sqrtf(A) * al;
      B0 = A * (Ap1 + Am1 * cw + tsa);
      B1 = -2.f * A * (Am1 + Ap1 * cw);
      B2 = A * (Ap1 + Am1 * cw - tsa);
      A0 = Ap1 - Am1 * cw + tsa;
      A1 = 2.f * (Am1 - Ap1 * cw);
      A2 = Ap1 - Am1 * cw - tsa;
    } else {                         // peaking
      float aA = al * A, aiA = al / A;
      B0 = 1.f + aA;  B1 = -2.f * cw; B2 = 1.f - aA;
      A0 = 1.f + aiA; A1 = -2.f * cw; A2 = 1.f - aiA;
    }
    float inv = 1.f / A0;
    cf[0][lane] = B0 * inv; cf[1][lane] = B1 * inv; cf[2][lane] = B2 * inv;
    cf[3][lane] = A1 * inv; cf[4][lane] = A2 * inv;
  }
  __syncthreads();

  float b0[NFILT], b1[NFILT], b2[NFILT], a1[NFILT], a2[NFILT];
#pragma unroll
  for (int k = 0; k < NFILT; k++) {
    b0[k] = cf[0][k]; b1[k] = cf[1][k]; b2[k] = cf[2][k];
    a1[k] = cf[3][k]; a2[k] = cf[4][k];
  }

  if (lane > STATE) return;  // lanes 0..20 run simulations

  float s1[NFILT] = {}, s2[NFILT] = {};
  if (lane >= 1) {           // unit initial state e_{lane-1}
    int i = lane - 1;
#pragma unroll
    for (int k = 0; k < NFILT; k++) {
      if (i == k)         s1[k] = 1.f;
      if (i == k + NFILT) s2[k] = 1.f;
    }
  }

  float* wch = ws + (size_t)ch * CH_STRIDE;
  for (int t = 0; t < BLK; t++) {
    float x = (lane == 0 && t == 0) ? 1.f : 0.f;
#pragma unroll
    for (int k = 0; k < NFILT; k++) {     // exact DF2T cascade step
      float y = b0[k] * x + s1[k];
      s1[k] = b1[k] * x - a1[k] * y + s2[k];
      s2[k] = b2[k] * x - a2[k] * y;
      x = y;
    }
    if (lane == 0) {
      wch[OFF_H + t] = x;                 // h[t]
      // G[:, BLK-1-t] = A^t * B  (state after processing impulse)
#pragma unroll
      for (int k = 0; k < NFILT; k++) {
        wch[OFF_G + k * BLK + (BLK - 1 - t)]           = s1[k];
        wch[OFF_G + (k + NFILT) * BLK + (BLK - 1 - t)] = s2[k];
      }
    } else {
      wch[OFF_O + t * STATE + (lane - 1)] = x;  // O[t][i] = C A^t e_i
    }
  }
  if (lane >= 1) {                         // T[:, i] = A^BLK e_i
#pragma unroll
    for (int k = 0; k < NFILT; k++) {
      wch[OFF_T + k * STATE + (lane - 1)]           = s1[k];
      wch[OFF_T + (k + NFILT) * STATE + (lane - 1)] = s2[k];
    }
  }
}

// ---------------------------------------------------------------------------
// Kernel 2: zero-state block responses  Y0 = H (lower-tri Toeplitz) @ X
// One wave handles a 16x32 output patch (2 column tiles sharing A fragments).
// Impulse response stored REVERSED + zero-padded in LDS so each A fragment is
// an ascending {q0,q0+1} pair -> single ds_load_2addr_b32, no swap movs, no
// predication, EXEC all-ones throughout (WMMA requirement).
// ---------------------------------------------------------------------------
__global__ void eq_fir_wmma(const float* __restrict__ x,
                            const float* __restrict__ ws,
                            float* __restrict__ y) {
  const int ct = blockIdx.x;   // pair of column tiles, 0..NBLK/32-1
  const int rt = blockIdx.y;   // row tile (t within block), 0..BLK/16-1
  const int ch = blockIdx.z;
  const int lane = threadIdx.x;

  // hr[q] = h[BLK-1-q] for q < BLK, 0 for q >= BLK (triangular edge)
  __shared__ float hr[BLK + HPAD];
  const float* wch = ws + (size_t)ch * CH_STRIDE;
  for (int i = lane; i < BLK + HPAD; i += 32)
    hr[i] = (i < BLK) ? wch[OFF_H + (BLK - 1 - i)] : 0.f;
  __syncthreads();

  const float* xch = x + (size_t)ch * LSIG;
  const int n  = lane & 15;    // M (for A) / N (for B) index
  const int hi = lane >> 4;    // k-pair select (ISA 16x4 / 4x16 layouts)
  const int j0 = ct * 32 + n;  // first signal-block column
  const int j1 = j0 + 16;      // second signal-block column
  const int m  = rt * 16 + n;  // output row within block

  const float* xp0 = xch + (size_t)j0 * BLK;
  const float* xp1 = xch + (size_t)j1 * BLK;

  v8f acc0 = {};
  v8f acc1 = {};
  const int kchunks = (rt + 1) * 4;   // triangular: tau <= t only
  for (int kk = 0; kk < kchunks; kk++) {
    const int k0 = kk * 4 + hi * 2;
    // A fragment: {h[m-k0], h[m-k0-1]} = {hr[q0], hr[q0+1]}, ascending pair
    const int q0 = (BLK - 1) - m + k0;
    v2f a;
    a.x = hr[q0];
    a.y = hr[q0 + 1];
    // B fragments: contiguous pair x[j*BLK + k0 .. +1]
    v2f b0 = *(const v2f*)(xp0 + k0);
    v2f b1 = *(const v2f*)(xp1 + k0);
    __builtin_prefetch(xp0 + k0 + 64, 0, 1);
    acc0 = __builtin_amdgcn_wmma_f32_16x16x4_f32(
        false, a, false, b0, (short)0, acc0, false, false);
    acc1 = __builtin_amdgcn_wmma_f32_16x16x4_f32(
        false, a, false, b1, (short)0, acc1, false, false);
  }
  // D layout: vgpr r -> row rt*16 + hi*8 + r; per lane 8 contiguous floats
  const int tbase = rt * 16 + hi * 8;
  float* yp0 = y + (size_t)ch * LSIG + (size_t)j0 * BLK + tbase;
  float* yp1 = y + (size_t)ch * LSIG + (size_t)j1 * BLK + tbase;
  *(v8f*)yp0 = acc0;
  *(v8f*)yp1 = acc1;
}

// ---------------------------------------------------------------------------
// Kernel 3: serial 20-dim state scan per channel (512 steps) + y += O*s.
// One wave per channel; G/O/T cached in LDS.
// ---------------------------------------------------------------------------
__global__ void eq_state(const float* __restrict__ x,
                         const float* __restrict__ ws,
                         float* __restrict__ y) {
  const int ch = blockIdx.x;
  const int lane = threadIdx.x;
  const float* wch = ws + (size_t)ch * CH_STRIDE;

  __shared__ float Gl[STATE * BLK];
  __shared__ float Ol[BLK * STATE];
  __shared__ float Tl[STATE * STATE];
  __shared__ float sst[STATE];
  __shared__ float snew[STATE];
  __shared__ float xl[BLK];

  for (int i = lane; i < STATE * BLK; i += 32) Gl[i] = wch[OFF_G + i];
  for (int i = lane; i < BLK * STATE; i += 32) Ol[i] = wch[OFF_O + i];
  for (int i = lane; i < STATE * STATE; i += 32) Tl[i] = wch[OFF_T + i];
  if (lane < STATE) sst[lane] = 0.f;
  __syncthreads();

  const float* xch = x + (size_t)ch * LSIG;
  float* ych = y + (size_t)ch * LSIG;

  for (int jb = 0; jb < NBLK; jb++) {
#pragma unroll
    for (int q = 0; q < 4; q++) xl[lane * 4 + q] = xch[jb * BLK + lane * 4 + q];
    __syncthreads();
    // y correction uses state at block START: y[t] += O[t][:] . s
    float yc[4];
#pragma unroll
    for (int q = 0; q < 4; q++) {
      const int t = lane * 4 + q;
      float acc = 0.f;
#pragma unroll
      for (int i = 0; i < STATE; i++) acc += Ol[t * STATE + i] * sst[i];
      yc[q] = acc;
    }
#pragma unroll
    for (int q = 0; q < 4; q++) ych[jb * BLK + lane * 4 + q] += yc[q];
    // s' = T s + G x_block
    if (lane < STATE) {
      float u = 0.f;
#pragma unroll 8
      for (int q = 0; q < BLK; q++) u += Gl[lane * BLK + q] * xl[q];
      float ts = 0.f;
#pragma unroll
      for (int i = 0; i < STATE; i++) ts += Tl[lane * STATE + i] * sst[i];
      snew[lane] = ts + u;
    }
    __syncthreads();
    if (lane < STATE) sst[lane] = snew[lane];
    __syncthreads();
  }
}

// ---------------------------------------------------------------------------
extern "C" void kernel_launch(void* const* d_in, const int* in_sizes, int n_in,
                              void* d_out, int out_size, void* d_ws, size_t ws_size,
                              hipStream_t stream) {
  (void)in_sizes; (void)n_in; (void)out_size; (void)ws_size;
  const float* xin = (const float*)d_in[0];   // (256,1,65536)
  const float* w0  = (const float*)d_in[1];   // (256,1,10)
  const float* qi  = (const float*)d_in[2];
  const float* lg  = (const float*)d_in[3];
  float* yout = (float*)d_out;                // (256,1,65536) f32
  float* ws   = (float*)d_ws;                 // >= 256*6144*4 = 6 MB

  eq_setup<<<dim3(256), dim3(32), 0, stream>>>(w0, qi, lg, ws);
  eq_fir_wmma<<<dim3(NBLK / 32, BLK / 16, 256), dim3(32), 0, stream>>>(xin, ws, yout);
  eq_state<<<dim3(256), dim3(32), 0, stream>>>(xin, ws, yout);
}